// ASSM_Illumination_48773648613388
// MI455X (gfx1250) — compile-verified
//
#include <hip/hip_runtime.h>
#include <hip/hip_bf16.h>
#include <math.h>

typedef float v2f __attribute__((ext_vector_type(2)));
typedef float v8f __attribute__((ext_vector_type(8)));

#define DIMC 192
#define HH   256
#define WW   256
#define LTOK 65536
#define C1   48
#define K2   432   /* 48*9 */
#define HID  96
#define NT   8

static __device__ __forceinline__ v8f wmma4(v2f a, v2f b, v8f c) {
  // D(16x16,f32) = A(16x4,f32) x B(4x16,f32) + C
  return __builtin_amdgcn_wmma_f32_16x16x4_f32(
      /*neg_a=*/false, a, /*neg_b=*/false, b,
      /*c_mod=*/(short)0, c, /*reuse_a=*/false, /*reuse_b=*/false);
}

// ---------------- kernel 1: luminance ----------------
__global__ void k_intensity(const float* __restrict__ x, float* __restrict__ intens) {
  int i = blockIdx.x * blockDim.x + threadIdx.x;
  if (i < LTOK) {
    const float* p = x + (size_t)i * DIMC;
    intens[i] = 0.299f * p[0] + 0.587f * p[1] + 0.114f * p[2];
  }
}

// ---------------- kernel 2: conv1 (1->48, 3x3 SAME) + LeakyReLU(0.2) ----------------
__global__ void k_conv1(const float* __restrict__ intens,
                        const float* __restrict__ w, const float* __restrict__ b,
                        float* __restrict__ f1) {
  int i = blockIdx.x * blockDim.x + threadIdx.x;
  if (i >= LTOK) return;
  int y = i >> 8, xc = i & 255;
  float nb[9];
#pragma unroll
  for (int ky = 0; ky < 3; ++ky)
#pragma unroll
    for (int kx = 0; kx < 3; ++kx) {
      int yy = y + ky - 1, xx = xc + kx - 1;
      nb[ky * 3 + kx] = (yy >= 0 && yy < HH && xx >= 0 && xx < WW) ? intens[yy * WW + xx] : 0.f;
    }
  for (int oc = 0; oc < C1; ++oc) {
    float s = b[oc];
#pragma unroll
    for (int t = 0; t < 9; ++t) s += w[oc * 9 + t] * nb[t];
    s = s > 0.f ? s : 0.2f * s;
    f1[(size_t)oc * LTOK + i] = s;  // planar layout, coalesced across pixels
  }
}

// ---------------- kernel 3: conv2(WMMA) + route MLP(WMMA) + softmax + gumbel ----------------
__global__ __launch_bounds__(128) void k_route(
    const float* __restrict__ f1, const float* __restrict__ x,
    const float* __restrict__ w2, const float* __restrict__ b2,
    const float* __restrict__ w1r, const float* __restrict__ b1r,
    const float* __restrict__ w2r, const float* __restrict__ b2r,
    const float* __restrict__ gum, const float* __restrict__ intens,
    const float* __restrict__ emb,
    float* __restrict__ gammaA, float* __restrict__ betaA) {
  __shared__ float patch[K2][16];     // im2col: 27.6 KB
  __shared__ float zt[16 * DIMC];     // z tile: 12.3 KB
  __shared__ float hm[16 * HID];      // hmid:   6.1 KB
  __shared__ float lg[16][NT];

  const int tid = threadIdx.x;
  const int lane = tid & 31, wv = tid >> 5;
  const int m16 = lane & 15, half = lane >> 4;
  const int blk = blockIdx.x;
  const int y = blk >> 4, xb = (blk & 15) << 4;
  const int tb = blk << 4;  // token base (= y*256 + xb)

  // --- gather zero-padded 3x3 patches of f1: patch[k = c*9+ky*3+kx][pixel] ---
  for (int idx = tid; idx < K2 * 16; idx += 128) {
    int k = idx >> 4, p = idx & 15;
    int c = k / 9, r = k % 9;
    int yy = y + r / 3 - 1, xx = xb + p + r % 3 - 1;
    patch[k][p] = (yy >= 0 && yy < HH && xx >= 0 && xx < WW)
                      ? f1[(size_t)c * LTOK + yy * WW + xx] : 0.f;
  }
  __syncthreads();

  // --- conv2 implicit GEMM: M=outchan(192 -> 12 tiles, 3/wave), N=16 pixels, K=432 ---
  {
    v8f a0c = {}, a1c = {}, a2c = {};
    const int oc0 = wv << 4, oc1 = (wv + 4) << 4, oc2 = (wv + 8) << 4;
    for (int k0 = 0; k0 < K2; k0 += 4) {
      int k = k0 + (half << 1);
      v2f bf; bf.x = patch[k][m16]; bf.y = patch[k + 1][m16];
      v2f a0, a1, a2;
      a0.x = w2[(oc0 + m16) * K2 + k]; a0.y = w2[(oc0 + m16) * K2 + k + 1];
      a1.x = w2[(oc1 + m16) * K2 + k]; a1.y = w2[(oc1 + m16) * K2 + k + 1];
      a2.x = w2[(oc2 + m16) * K2 + k]; a2.y = w2[(oc2 + m16) * K2 + k + 1];
      a0c = wmma4(a0, bf, a0c);
      a1c = wmma4(a1, bf, a1c);
      a2c = wmma4(a2, bf, a2c);
    }
#pragma unroll
    for (int r = 0; r < 8; ++r) {  // D: M = r + 8*half (outchan), N = m16 (pixel)
      zt[m16 * DIMC + oc0 + r + (half << 3)] = a0c[r];
      zt[m16 * DIMC + oc1 + r + (half << 3)] = a1c[r];
      zt[m16 * DIMC + oc2 + r + (half << 3)] = a2c[r];
    }
  }
  __syncthreads();

  // --- z = x + 0.3*(conv2 + b2) ---
  for (int idx = tid; idx < 16 * DIMC; idx += 128) {
    int p = idx / DIMC, c = idx % DIMC;
    zt[idx] = x[(size_t)(tb + p) * DIMC + c] + 0.3f * (zt[idx] + b2[c]);
  }
  __syncthreads();

  // --- route1: hmid = gelu(z @ W1^T + b1): M=16 tokens, N=96 (6 tiles), K=192 ---
  for (int t = wv; t < 6; t += 4) {
    int nb_ = t << 4;
    v8f acc = {};
    for (int k0 = 0; k0 < DIMC; k0 += 4) {
      int k = k0 + (half << 1);
      v2f a; a.x = zt[m16 * DIMC + k];            a.y = zt[m16 * DIMC + k + 1];
      v2f b; b.x = w1r[(nb_ + m16) * DIMC + k];   b.y = w1r[(nb_ + m16) * DIMC + k + 1];
      acc = wmma4(a, b, acc);
    }
#pragma unroll
    for (int r = 0; r < 8; ++r) {  // D: M = token, N = hidden col
      int tok = r + (half << 3), col = nb_ + m16;
      float v = acc[r] + b1r[col];
      v = 0.5f * v * (1.0f + erff(v * 0.70710678118654752f));  // exact GELU
      hm[tok * HID + col] = v;
    }
  }
  __syncthreads();

  // --- route2: 16 tokens x 8 logits, K=96 (one (tok,o) pair per thread) ---
  {
    int tok = tid >> 3, o = tid & 7;
    float s = b2r[o];
    for (int k = 0; k < HID; ++k) s += hm[tok * HID + k] * w2r[o * HID + k];
    lg[tok][o] = s;
  }
  __syncthreads();

  // --- softmax + illum_sim + gumbel argmax -> gamma/beta (one thread per token) ---
  if (tid < 16) {
    int tok = tid;
    float mx = lg[tok][0];
#pragma unroll
    for (int j = 1; j < NT; ++j) mx = fmaxf(mx, lg[tok][j]);
    float rw[NT], ssum = 0.f;
#pragma unroll
    for (int j = 0; j < NT; ++j) { rw[j] = expf(lg[tok][j] - mx); ssum += rw[j]; }
    float inv = 1.f / ssum;
    float it = intens[tb + tok];
    int kbest = 0; float best = -1e30f;
#pragma unroll
    for (int j = 0; j < NT; ++j) {
      rw[j] *= inv;
      float sim = 1.f - fabsf(emb[j] - it);
      float u = gum[(size_t)(tb + tok) * NT + j];
      u = fminf(fmaxf(u, 1e-10f), 1.0f);
      float g = -logf(-logf(u));
      float v = rw[j] * sim + g;                 // argmax(softmax(z)) == argmax(z)
      if (v > best) { best = v; kbest = j; }
    }
    float p0 = emb[kbest];                        // cls_policy @ illum_embedding
    gammaA[tb + tok] = 0.3f + 0.7f / (1.f + expf(-p0));
    betaA[tb + tok]  = -0.5f + rw[0];             // prompt[...,1] = route_weights[...,0]
  }
}

// ---------------- kernel 4: gate + D + LayerNorm + out-projection (WMMA) ----------------
__global__ __launch_bounds__(128) void k_out(
    const float* __restrict__ x, const float* __restrict__ gammaA,
    const float* __restrict__ betaA, const float* __restrict__ Dv,
    const float* __restrict__ lnw, const float* __restrict__ lnb,
    const float* __restrict__ wout, const float* __restrict__ bout,
    float* __restrict__ out) {
  __shared__ float yn[16 * DIMC];
  __shared__ float mu_s[16], rs_s[16];
  const int tid = threadIdx.x;
  const int lane = tid & 31, wv = tid >> 5;
  const int m16 = lane & 15, half = lane >> 4;
  const int tb = blockIdx.x << 4;

  // y[t,c] = D_c * (gamma_t * x[t,c] + beta_t)   (scan scalar cancels in LN exactly)
  for (int idx = tid; idx < 16 * DIMC; idx += 128) {
    int p = idx / DIMC, c = idx % DIMC;
    yn[idx] = Dv[c] * (gammaA[tb + p] * x[(size_t)(tb + p) * DIMC + c] + betaA[tb + p]);
  }
  __syncthreads();
  if (tid < 16) {
    float s = 0.f, s2 = 0.f;
    for (int c = 0; c < DIMC; ++c) { float v = yn[tid * DIMC + c]; s += v; s2 += v * v; }
    float mu = s / DIMC;
    mu_s[tid] = mu;
    rs_s[tid] = rsqrtf(s2 / DIMC - mu * mu + 1e-5f);
  }
  __syncthreads();
  for (int idx = tid; idx < 16 * DIMC; idx += 128) {
    int p = idx / DIMC, c = idx % DIMC;
    yn[idx] = (yn[idx] - mu_s[p]) * rs_s[p] * lnw[c] + lnb[c];
  }
  __syncthreads();

  // out = yn @ out_w^T + out_b : M=16 tokens, N=192 (12 tiles, 3/wave), K=192
  for (int ti = 0; ti < 3; ++ti) {
    int nb_ = (wv + 4 * ti) << 4;
    v8f acc = {};
    for (int k0 = 0; k0 < DIMC; k0 += 4) {
      int k = k0 + (half << 1);
      v2f a; a.x = yn[m16 * DIMC + k];            a.y = yn[m16 * DIMC + k + 1];
      v2f b; b.x = wout[(nb_ + m16) * DIMC + k];  b.y = wout[(nb_ + m16) * DIMC + k + 1];
      acc = wmma4(a, b, acc);
    }
#pragma unroll
    for (int r = 0; r < 8; ++r) {  // D: M = token, N = out channel (64B/half-wave coalesced)
      int tok = tb + r + (half << 3), o = nb_ + m16;
      out[(size_t)tok * DIMC + o] = acc[r] + bout[o];
    }
  }
}

extern "C" void kernel_launch(void* const* d_in, const int* in_sizes, int n_in,
                              void* d_out, int out_size, void* d_ws, size_t ws_size,
                              hipStream_t stream) {
  (void)in_sizes; (void)n_in; (void)out_size; (void)ws_size;
  const float* x   = (const float*)d_in[0];
  const float* gum = (const float*)d_in[1];
  const float* c1w = (const float*)d_in[2];
  const float* c1b = (const float*)d_in[3];
  const float* c2w = (const float*)d_in[4];
  const float* c2b = (const float*)d_in[5];
  const float* r1w = (const float*)d_in[6];
  const float* r1b = (const float*)d_in[7];
  const float* r2w = (const float*)d_in[8];
  const float* r2b = (const float*)d_in[9];
  // d_in[10] = A_log (dead: cancels in LayerNorm), d_in[11] = D
  const float* Dv  = (const float*)d_in[11];
  const float* lnw = (const float*)d_in[12];
  const float* lnb = (const float*)d_in[13];
  const float* ow  = (const float*)d_in[14];
  const float* ob  = (const float*)d_in[15];
  const float* emb = (const float*)d_in[16];
  float* out = (float*)d_out;

  // workspace layout (floats): intens[L] | f1[48*L] | gamma[L] | beta[L]  (~13.4 MB)
  float* ws     = (float*)d_ws;
  float* intens = ws;
  float* f1     = ws + LTOK;
  float* gammaA = f1 + (size_t)C1 * LTOK;
  float* betaA  = gammaA + LTOK;

  k_intensity<<<LTOK / 256, 256, 0, stream>>>(x, intens);
  k_conv1<<<LTOK / 256, 256, 0, stream>>>(intens, c1w, c1b, f1);
  k_route<<<LTOK / 16, 128, 0, stream>>>(f1, x, c2w, c2b, r1w, r1b, r2w, r2b,
                                         gum, intens, emb, gammaA, betaA);
  k_out<<<LTOK / 16, 128, 0, stream>>>(x, gammaA, betaA, Dv, lnw, lnb, ow, ob, out);
}